// phase_62285615727523
// MI455X (gfx1250) — compile-verified
//
#include <hip/hip_runtime.h>

// Pixel-unshuffle (2x2 space-to-depth), fp32, pure streaming permutation.
// x: (64,3,512,512) -> out: (4, 64,3,256,256); out[p,b,c,i,j] = x[b,c,2i+p/2,2j+p%2]
// Traffic: 402 MB total -> ~17.3 us floor @ 23.3 TB/s HBM. No FLOPs -> no WMMA;
// CDNA5 path used: async global->LDS data mover + s_wait_asynccnt + NT stores.

typedef __attribute__((ext_vector_type(4))) float v4f;     // for VGPR data
typedef int v4i __attribute__((vector_size(16)));          // builtin pointee type

#define B_   64
#define C_   3
#define H_   512
#define W_   512
#define BC_  (B_ * C_)        // 192
#define Hh_  (H_ / 2)         // 256
#define Wh_  (W_ / 2)         // 256
#define W8_  (W_ / 8)         // 64 8-float segments per input row
#define NT_TOTAL (B_ * C_ * H_ * W8_)   // 6,291,456 threads
#define PLANE_ (BC_ * Hh_ * Wh_)        // elems per phase = 12,582,912

#if __has_builtin(__builtin_amdgcn_global_load_async_to_lds_b128)
#define HAVE_ASYNC_LDS 1
#else
#define HAVE_ASYNC_LDS 0
#endif

#if __has_builtin(__builtin_amdgcn_s_wait_asynccnt)
#define WAIT_ASYNCCNT0() __builtin_amdgcn_s_wait_asynccnt(0)
#else
#define WAIT_ASYNCCNT0() asm volatile("s_wait_asynccnt 0" ::: "memory")
#endif

extern "C" __global__ void __launch_bounds__(256)
pixel_unshuffle_2x2(const float* __restrict__ x, float* __restrict__ out)
{
    const unsigned tid = threadIdx.x;
    const unsigned t   = blockIdx.x * 256u + tid;

    // t maps linearly onto the input: in_off = t*8 (coalesced streaming scan)
    const unsigned w8   = t & (W8_ - 1);          // segment within row
    const unsigned rest = t >> 6;
    const unsigned h    = rest & (H_ - 1);        // input row
    const unsigned bc   = rest >> 9;              // b*C + c
    const unsigned i    = h >> 1;                 // output row
    const unsigned dy   = h & 1;                  // row phase
    const unsigned j0   = w8 * 4;                 // first output col of this thread

    // phase p = 2*dy + dz; dz=0 (even cols) and dz=1 (odd cols) are PLANE_ apart
    const unsigned outE = (((2u * dy) * BC_ + bc) * Hh_ + i) * Wh_ + j0;
    const unsigned outO = outE + PLANE_;

    v4f a, b;
#if HAVE_ASYNC_LDS
    // Stage each lane's 32B through its private LDS slot via the CDNA5 async
    // data mover. Lane-private round trip -> no barrier, only s_wait_asynccnt.
    __shared__ __align__(16) float smem[256 * 8];   // 8 KB / block
    __attribute__((address_space(1))) v4i* g =
        (__attribute__((address_space(1))) v4i*)((float*)x + (size_t)t * 8u);
    __attribute__((address_space(3))) v4i* l =
        (__attribute__((address_space(3))) v4i*)(&smem[tid * 8u]);
    __builtin_amdgcn_global_load_async_to_lds_b128(g, l, 0,  0);
    __builtin_amdgcn_global_load_async_to_lds_b128(g + 1, l + 1, 0, 0);
    WAIT_ASYNCCNT0();
    const __attribute__((address_space(3))) v4f* lv =
        (const __attribute__((address_space(3))) v4f*)(&smem[tid * 8u]);
    a = lv[0];            // ds_load_b128
    b = lv[1];
#else
    // Direct path: two NT b128 loads (wave32 covers 1KB contiguous)
    const v4f* xin = (const v4f*)(x + (size_t)t * 8u);
    a = __builtin_nontemporal_load(xin);
    b = __builtin_nontemporal_load(xin + 1);
#endif

    // De-interleave: even cols -> phase (dy,0), odd cols -> phase (dy,1)
    v4f e = { a.x, a.z, b.x, b.z };
    v4f o = { a.y, a.w, b.y, b.w };

    // 16B NT stores; each wave writes 512B contiguous per phase
    __builtin_nontemporal_store(e, (v4f*)(out + outE));
    __builtin_nontemporal_store(o, (v4f*)(out + outO));
}

extern "C" void kernel_launch(void* const* d_in, const int* in_sizes, int n_in,
                              void* d_out, int out_size, void* d_ws, size_t ws_size,
                              hipStream_t stream) {
    const float* x = (const float*)d_in[0];
    float* out     = (float*)d_out;
    (void)in_sizes; (void)n_in; (void)out_size; (void)d_ws; (void)ws_size;

    dim3 grid(NT_TOTAL / 256);   // 24,576 blocks
    dim3 block(256);             // 8 wave32 per block
    pixel_unshuffle_2x2<<<grid, block, 0, stream>>>(x, out);
}